// NoBlankCTC_29222957482465
// MI455X (gfx1250) — compile-verified
//
#include <hip/hip_runtime.h>
#include <math.h>

// Problem constants (reference: T,B,V,L = 1024,32,1000,256)
#define T_  1024
#define B_  32
#define V_  1000
#define L_  256
#define NEGV (-1e13f)
#define ROWS_PER_BLK 16   // rows of yseq handled per workgroup in k_lse
#define PD 8              // gather prefetch depth in k_scan

typedef __attribute__((ext_vector_type(2))) float        v2f;
typedef __attribute__((ext_vector_type(8))) float        v8f;
typedef __attribute__((ext_vector_type(4))) unsigned int u32x4;
typedef __attribute__((ext_vector_type(8))) int          i32x8;
typedef __attribute__((ext_vector_type(4))) int          i32x4;

#if __has_builtin(__builtin_amdgcn_tensor_load_to_lds) && \
    __has_builtin(__builtin_amdgcn_s_wait_tensorcnt)
#define USE_TDM 1
#endif

// ---------------------------------------------------------------------------
// Kernel 1: Z[r] = logsumexp_v(yseq[r, :]) for r = t*B + b  (32768 rows)
// 128 threads (4 waves) per block, 16 rows per block.
// Staging: TDM TENSOR_LOAD_TO_LDS moves the 16x1000 fp32 tile (64 KB) into
// LDS via the DMA engine (D# built per ISA sec 8.3/8.4), waited with
// s_wait_tensorcnt. Row-sum of exp(x-max) runs on V_WMMA_F32_16X16X4_F32
// with two independent accumulators to avoid the C->C RAW chain.
// ---------------------------------------------------------------------------
__global__ __launch_bounds__(128)
void k_lse(const float* __restrict__ yseq, float* __restrict__ Z) {
  __shared__ __align__(16) float rowbuf[ROWS_PER_BLK * V_];  // 64000 B
  __shared__ float redmax[ROWS_PER_BLK * 8];
  __shared__ float rowmax[ROWS_PER_BLK];
  __shared__ float wpart[4 * ROWS_PER_BLK];

  const int tid  = threadIdx.x;
  const int lane = tid & 31;
  const int wave = tid >> 5;
  const size_t r0 = (size_t)blockIdx.x * ROWS_PER_BLK;

#ifdef USE_TDM
  // ---- stage 16 rows (16000 fp32 = 64000 B) via Tensor Data Mover ----
  if (wave == 0) {
    const unsigned long long ga =
        (unsigned long long)(const void*)(yseq + r0 * V_);
    const unsigned int lds = (unsigned int)(size_t)(void*)rowbuf; // addr[31:0] = LDS byte addr
    u32x4 g0;
    g0[0] = 1u;                                  // count=1, is_restore=0, gather off
    g0[1] = lds;                                 // lds_addr
    g0[2] = (unsigned int)(ga & 0xFFFFFFFFull);  // global_addr[31:0]
    g0[3] = (unsigned int)((ga >> 32) & 0x1FFFFFFull)  // global_addr[56:32]
          | (2u << 30);                          // type = 2 ("image")
    i32x8 g1;
    g1[0] = (int)(2u << 16);          // wg_mask=0 | data_size=2 (4B) | no flags
    g1[1] = (int)(1000u << 16);       // atomic_barrier=0 | tensor_dim0[15:0]=1000
    g1[2] = (int)(16u << 16);         // tensor_dim0[31:16]=0 | tensor_dim1[15:0]=16
    g1[3] = (int)(1000u << 16);       // tensor_dim1[31:16]=0 | tile_dim0=1000
    g1[4] = 16;                       // tile_dim1=16 | tile_dim2=0
    g1[5] = 1000;                     // tensor_dim0_stride[31:0]=1000
    g1[6] = 0;                        // stride0[47:32]=0 | stride1[15:0]=0
    g1[7] = 0;                        // stride1[47:16]=0
    i32x4 gz;  gz[0] = 0; gz[1] = 0; gz[2] = 0; gz[3] = 0;
    i32x8 gz8; gz8[0] = 0; gz8[1] = 0; gz8[2] = 0; gz8[3] = 0;
               gz8[4] = 0; gz8[5] = 0; gz8[6] = 0; gz8[7] = 0;
    // 6-arg toolchain form: (g0, g1, g2, g3, g4, cpol)
    __builtin_amdgcn_tensor_load_to_lds(g0, g1, gz, gz, gz8, 0);
    __builtin_amdgcn_s_wait_tensorcnt(0);
  }
  __syncthreads();
#else
  // ---- fallback: cooperative coalesced b128 copy ----
  const float4* __restrict__ src4 =
      reinterpret_cast<const float4*>(yseq + r0 * V_);
  float4* dst4 = reinterpret_cast<float4*>(rowbuf);
  for (int i = tid; i < (ROWS_PER_BLK * V_) / 4; i += 128) {
    dst4[i] = src4[i];
  }
  __syncthreads();
#endif

  // ---- per-row max (8 threads per row, then 16-thread combine) ----
  {
    const int m = tid >> 3;          // row 0..15
    const int j = tid & 7;
    float mx = -3.0e38f;
    for (int k = j; k < V_; k += 8) mx = fmaxf(mx, rowbuf[m * V_ + k]);
    redmax[m * 8 + j] = mx;
  }
  __syncthreads();
  if (tid < ROWS_PER_BLK) {
    float mx = redmax[tid * 8];
#pragma unroll
    for (int j = 1; j < 8; ++j) mx = fmaxf(mx, redmax[tid * 8 + j]);
    rowmax[tid] = mx;
  }
  __syncthreads();

  // ---- sum(exp(x - max)) via WMMA f32 16x16x4, 2 accumulators ----
  // ISA A-layout (32-bit A 16x4): lanes 0-15 -> M=lane, K = kb+0 (v0), kb+1 (v1)
  //                               lanes16-31 -> M=lane-16, K = kb+2, kb+3
  const int   m    = lane & 15;
  const int   ko   = (lane < 16) ? 0 : 2;
  const float mrow = rowmax[m];
  v2f bones; bones.x = 1.0f; bones.y = 1.0f;   // B = all ones (4x16)
  v8f acc0 = {}, acc1 = {};
  // 250 K-steps of 4; wave w owns step pairs s = 2w + 8k  (exact tiling, no tail)
  for (int s = wave * 2; s < V_ / 4; s += 8) {
    const int kb0 = s * 4;
    v2f a0, a1;
    a0.x = __expf(rowbuf[m * V_ + kb0 + ko]     - mrow);
    a0.y = __expf(rowbuf[m * V_ + kb0 + ko + 1] - mrow);
    a1.x = __expf(rowbuf[m * V_ + kb0 + 4 + ko]     - mrow);
    a1.y = __expf(rowbuf[m * V_ + kb0 + 4 + ko + 1] - mrow);
    acc0 = __builtin_amdgcn_wmma_f32_16x16x4_f32(
        false, a0, false, bones, (short)0, acc0, false, false);
    acc1 = __builtin_amdgcn_wmma_f32_16x16x4_f32(
        false, a1, false, bones, (short)0, acc1, false, false);
  }

  // D layout (16x16 f32): lane 0 vgpr i -> (M=i, N=0); lane 16 vgpr i -> (M=8+i, N=0)
  if (lane == 0 || lane == 16) {
    const int half = (lane == 16) ? 8 : 0;
#pragma unroll
    for (int i = 0; i < 8; ++i)
      wpart[wave * ROWS_PER_BLK + half + i] = acc0[i] + acc1[i];
  }
  __syncthreads();
  if (tid < ROWS_PER_BLK) {
    float s = wpart[tid] + wpart[16 + tid] + wpart[32 + tid] + wpart[48 + tid];
    Z[r0 + tid] = rowmax[tid] + __logf(s);
  }
}

// ---------------------------------------------------------------------------
// Kernel 2: per-batch CTC forward scan. One workgroup per b, thread = l.
// alpha[t,l] = logaddexp(alpha[t-1,l], alpha[t-1,l-1]) masked + em[t,l],
// em[t,l] = yseq[t,b,label[b,l]] - Z[t,b]  (gathers hit L2: yseq fits in 192MB)
// Gathers software-pipelined PD deep; shift-by-one via LDS + barriers.
// ---------------------------------------------------------------------------
__global__ __launch_bounds__(L_)
void k_scan(const float* __restrict__ yseq, const int* __restrict__ label,
            const int* __restrict__ ilen, const int* __restrict__ tlen,
            const float* __restrict__ Z, float* __restrict__ sel) {
  __shared__ float abuf[L_];
  const int b = blockIdx.x;
  const int l = threadIdx.x;

  const int  lab     = label[b * L_ + l];
  const int  il      = ilen[b];
  const int  tl      = tlen[b];
  const bool outside = (l >= tl);
  const float* __restrict__ gcol = yseq + (size_t)b * V_ + lab;  // + t*B*V

  float cur[PD];
#pragma unroll
  for (int i = 0; i < PD; ++i) cur[i] = gcol[(size_t)i * (B_ * V_)];

  float alpha = (l == 0) ? 0.0f : NEGV;   // init state before t=0

  for (int tb = 0; tb < T_; tb += PD) {
    // issue next PD gathers early (consumed next outer iteration)
    float nxt[PD];
#pragma unroll
    for (int i = 0; i < PD; ++i) {
      const int tn = tb + PD + i;
      nxt[i] = (tn < T_) ? gcol[(size_t)tn * (B_ * V_)] : 0.0f;
    }
#pragma unroll
    for (int i = 0; i < PD; ++i) {
      const int t = tb + i;
      const float z = Z[t * B_ + b];            // uniform -> scalar load
      float sh = NEGV;
      if (t > 0 && l > 0) sh = abuf[l - 1];     // alpha[t-1, l-1]
      __syncthreads();                          // all reads done before overwrite
      const float a  = alpha;
      const float mx = fmaxf(a, sh);
      float raw = mx + __logf(1.0f + __expf(-fabsf(a - sh)));  // logaddexp
      raw = outside ? NEGV : raw;
      alpha = raw + (cur[i] - z);
      abuf[l] = alpha;
      __syncthreads();                          // stores visible for next step
      if (t == il - 1 && l == tl - 1) sel[b] = alpha;
    }
#pragma unroll
    for (int i = 0; i < PD; ++i) cur[i] = nxt[i];
  }
}

// ---------------------------------------------------------------------------
// Kernel 3: out = -mean_b sel[b]
// ---------------------------------------------------------------------------
__global__ void k_final(const float* __restrict__ sel, float* __restrict__ out) {
  if (threadIdx.x == 0) {
    float s = 0.0f;
    for (int b = 0; b < B_; ++b) s += sel[b];
    out[0] = -s / (float)B_;
  }
}

// ---------------------------------------------------------------------------
extern "C" void kernel_launch(void* const* d_in, const int* in_sizes, int n_in,
                              void* d_out, int out_size, void* d_ws, size_t ws_size,
                              hipStream_t stream) {
  (void)in_sizes; (void)n_in; (void)out_size; (void)ws_size;
  const float* yseq  = (const float*)d_in[0];   // [T,B,V] fp32
  const int*   label = (const int*)d_in[1];     // [B,L]
  const int*   ilen  = (const int*)d_in[2];     // [B]
  const int*   tlen  = (const int*)d_in[3];     // [B]

  float* Z   = (float*)d_ws;          // T*B floats = 128 KiB
  float* sel = Z + (size_t)T_ * B_;   // B floats

  k_lse  <<<(T_ * B_) / ROWS_PER_BLK, 128, 0, stream>>>(yseq, Z);
  k_scan <<<B_, L_, 0, stream>>>(yseq, label, ilen, tlen, Z, sel);
  k_final<<<1, 32, 0, stream>>>(sel, (float*)d_out);
}